// GRAPH_AUTOENCODER_52578989637996
// MI455X (gfx1250) — compile-verified
//
#include <hip/hip_runtime.h>
#include <math.h>

// Problem constants (from reference)
#define NN  50000
#define EE  400000
#define F0C 128
#define H0C 256
#define H1C 128
#define GC  500
#define GPC 1000

typedef float v2f __attribute__((ext_vector_type(2)));
typedef float v8f __attribute__((ext_vector_type(8)));

// ---------------------------------------------------------------------------
// GEMM: out = act(A(nxK) @ W(KxM) + bias), fp32 via V_WMMA_F32_16X16X4_F32.
// Block = 128 threads (4 waves). Block tile 64x64, wave tile 16x64.
// B tile stored K-transposed in LDS so every fragment is a contiguous K-pair
// (single ds_load_2addr offset1:1, no repacking movs in the WMMA loop).
// act: 0 = none, 1 = relu, 2 = sigmoid. bias may be nullptr.
// ---------------------------------------------------------------------------
__global__ __launch_bounds__(128) void gemm_wmma_k(
    const float* __restrict__ Amat, const float* __restrict__ Wmat,
    const float* __restrict__ bias, float* __restrict__ out,
    int n, int K, int M, int act)
{
  __shared__ float As[64][17];    // As[row][k]
  __shared__ float BsT[64][17];   // BsT[col][k]

  const int tid  = threadIdx.x;
  const int lane = tid & 31;
  const int wave = tid >> 5;
  const int r0 = blockIdx.y << 6;
  const int c0 = blockIdx.x << 6;

  const v8f zero = {0.f, 0.f, 0.f, 0.f, 0.f, 0.f, 0.f, 0.f};
  v8f acc0 = zero, acc1 = zero, acc2 = zero, acc3 = zero;

  const int m  = lane & 15;          // A row within tile / B,C,D column
  const int ko = (lane >> 4) << 1;   // 0 or 2: K sub-offset for this half-wave
  const int rb = wave << 4;          // wave's 16-row strip inside 64-row tile

  for (int kb = 0; kb < K; kb += 16) {
    // stage A tile (64x16), guarded rows; consecutive tids -> consecutive k
    for (int i = tid; i < 64 * 16; i += 128) {
      int r = i >> 4, k = i & 15;
      int gr = r0 + r;
      As[r][k] = (gr < n) ? Amat[(size_t)gr * K + (kb + k)] : 0.0f;
    }
    // stage B tile (16x64) transposed; consecutive tids -> consecutive cols
    for (int i = tid; i < 16 * 64; i += 128) {
      int r = i >> 6, c = i & 63;
      BsT[c][r] = Wmat[(size_t)(kb + r) * M + (c0 + c)];
    }
    __syncthreads();

#pragma unroll
    for (int kk = 0; kk < 16; kk += 4) {
      v2f a;
      a[0] = As[rb + m][kk + ko];
      a[1] = As[rb + m][kk + ko + 1];
      v2f b0, b1, b2, b3;
      b0[0] = BsT[ 0 + m][kk + ko]; b0[1] = BsT[ 0 + m][kk + ko + 1];
      b1[0] = BsT[16 + m][kk + ko]; b1[1] = BsT[16 + m][kk + ko + 1];
      b2[0] = BsT[32 + m][kk + ko]; b2[1] = BsT[32 + m][kk + ko + 1];
      b3[0] = BsT[48 + m][kk + ko]; b3[1] = BsT[48 + m][kk + ko + 1];
      acc0 = __builtin_amdgcn_wmma_f32_16x16x4_f32(false, a, false, b0, (short)0, acc0, false, false);
      acc1 = __builtin_amdgcn_wmma_f32_16x16x4_f32(false, a, false, b1, (short)0, acc1, false, false);
      acc2 = __builtin_amdgcn_wmma_f32_16x16x4_f32(false, a, false, b2, (short)0, acc2, false, false);
      acc3 = __builtin_amdgcn_wmma_f32_16x16x4_f32(false, a, false, b3, (short)0, acc3, false, false);
    }
    __syncthreads();
  }

  // Epilogue: C/D layout -> row = v + 8*(lane>=16), col = lane&15
  const int rowbase = r0 + rb + ((lane >> 4) << 3);
  v8f accv[4] = {acc0, acc1, acc2, acc3};
#pragma unroll
  for (int j = 0; j < 4; ++j) {
    const int col = c0 + j * 16 + m;
    const float bv = bias ? bias[col] : 0.0f;
#pragma unroll
    for (int v = 0; v < 8; ++v) {
      const int row = rowbase + v;
      if (row < n) {
        float y = accv[j][v] + bv;
        if (act == 1) y = fmaxf(y, 0.0f);
        else if (act == 2) y = 1.0f / (1.0f + expf(-y));
        out[(size_t)row * M + col] = y;
      }
    }
  }
}

// ---------------------------------------------------------------------------
// Degree kernels (deg = in-degree(dst) + 1 self loop)
// ---------------------------------------------------------------------------
__global__ void fill_k(float* __restrict__ p, float v, int n) {
  int i = blockIdx.x * blockDim.x + threadIdx.x;
  if (i < n) p[i] = v;
}
__global__ void deg_count_k(const int* __restrict__ dst, float* __restrict__ deg, int E) {
  int i = blockIdx.x * blockDim.x + threadIdx.x;
  if (i < E) atomicAdd(&deg[dst[i]], 1.0f);
}
// in: deg (overwritten with rsqrt(deg)); out: invd = 1/deg
__global__ void deg_fin_k(float* __restrict__ deg_dinv, float* __restrict__ invd, int n) {
  int i = blockIdx.x * blockDim.x + threadIdx.x;
  if (i < n) {
    float d = deg_dinv[i];
    deg_dinv[i] = rsqrtf(d);
    invd[i] = 1.0f / d;
  }
}

// ---------------------------------------------------------------------------
// GCN aggregation: out = self-loop term + bias, then scatter-add over edges
// ---------------------------------------------------------------------------
__global__ void agg_init_k(const float* __restrict__ h, const float* __restrict__ invd,
                           const float* __restrict__ bias, float* __restrict__ out,
                           int n, int mshift)
{
  const int M = 1 << mshift;
  long long base = ((long long)blockIdx.x * blockDim.x + threadIdx.x) << 2;
  if (base >= ((long long)n << mshift)) return;
  int r = (int)(base >> mshift);
  int c = (int)(base & (M - 1));
  float s = invd[r];
  float4 hv = *(const float4*)(h + base);
  float4 o;
  o.x = hv.x * s + bias[c + 0];
  o.y = hv.y * s + bias[c + 1];
  o.z = hv.z * s + bias[c + 2];
  o.w = hv.w * s + bias[c + 3];
  *(float4*)(out + base) = o;
}

// eshift = log2(M/4); one thread handles 4 channels of one edge
__global__ void scatter_k(const float* __restrict__ h, const int* __restrict__ src,
                          const int* __restrict__ dst, const float* __restrict__ dinv,
                          float* __restrict__ out, int E, int eshift, int M)
{
  long long tid = (long long)blockIdx.x * blockDim.x + threadIdx.x;
  int e = (int)(tid >> eshift);
  if (e >= E) return;
  int c4 = (((int)tid & ((1 << eshift) - 1)) << 2);
  int s = src[e], d = dst[e];
  float coef = dinv[s] * dinv[d];
  const float4 hv = *(const float4*)(h + (size_t)s * M + c4);
  float* o = out + (size_t)d * M + c4;
  atomicAdd(o + 0, hv.x * coef);
  atomicAdd(o + 1, hv.y * coef);
  atomicAdd(o + 2, hv.z * coef);
  atomicAdd(o + 3, hv.w * coef);
}

// ---------------------------------------------------------------------------
// BatchNorm (training stats, biased var)
// ---------------------------------------------------------------------------
__global__ __launch_bounds__(256) void bn_stats_k(const float* __restrict__ x, int n, int mshift,
                                                  int rows_per_block, float* __restrict__ sum,
                                                  float* __restrict__ ssq)
{
  __shared__ float ls[256], lq[256];
  const int M = 1 << mshift;
  const int t = threadIdx.x;
  const int c = t & (M - 1);
  const int rseg = t >> mshift;
  const int nseg = 256 >> mshift;
  int r0 = blockIdx.x * rows_per_block;
  int r1 = min(r0 + rows_per_block, n);
  float s = 0.f, q = 0.f;
  for (int r = r0 + rseg; r < r1; r += nseg) {
    float v = x[(size_t)r * M + c];
    s += v; q += v * v;
  }
  ls[t] = s; lq[t] = q;
  __syncthreads();
  if (t < M) {
    for (int k = M; k < 256; k += M) { s += ls[t + k]; q += lq[t + k]; }
    atomicAdd(&sum[c], s);
    atomicAdd(&ssq[c], q);
  }
}

__global__ void bn_coef_k(const float* __restrict__ sum, const float* __restrict__ ssq,
                          const float* __restrict__ g, const float* __restrict__ b,
                          int M, float invN, float* __restrict__ cA, float* __restrict__ cC)
{
  int c = blockIdx.x * blockDim.x + threadIdx.x;
  if (c >= M) return;
  float mean = sum[c] * invN;
  float var  = ssq[c] * invN - mean * mean;
  float a = (g ? g[c] : 1.0f) * rsqrtf(var + 1e-5f);
  cA[c] = a;
  cC[c] = (b ? b[c] : 0.0f) - mean * a;
}

__global__ void bn_apply_k(const float* __restrict__ x, const float* __restrict__ cA,
                           const float* __restrict__ cC, const float* __restrict__ res,
                           float* __restrict__ out, int n, int mshift, int do_relu)
{
  const int M = 1 << mshift;
  long long base = ((long long)blockIdx.x * blockDim.x + threadIdx.x) << 2;
  if (base >= ((long long)n << mshift)) return;
  int c = (int)(base & (M - 1));
  float4 xv = *(const float4*)(x + base);
  float4 y;
  y.x = xv.x * cA[c + 0] + cC[c + 0];
  y.y = xv.y * cA[c + 1] + cC[c + 1];
  y.z = xv.z * cA[c + 2] + cC[c + 2];
  y.w = xv.w * cA[c + 3] + cC[c + 3];
  if (res) {
    float4 rv = *(const float4*)(res + base);
    y.x += rv.x; y.y += rv.y; y.z += rv.z; y.w += rv.w;
  }
  if (do_relu) {
    y.x = fmaxf(y.x, 0.f); y.y = fmaxf(y.y, 0.f);
    y.z = fmaxf(y.z, 0.f); y.w = fmaxf(y.w, 0.f);
  }
  *(float4*)(out + base) = y;
}

// ---------------------------------------------------------------------------
// Row L2 normalize (M = 128 fixed): one wave per row
// ---------------------------------------------------------------------------
__global__ void l2norm_k(const float* __restrict__ in, float* __restrict__ out, int n)
{
  int wave = threadIdx.x >> 5, lane = threadIdx.x & 31;
  int row = blockIdx.x * (blockDim.x >> 5) + wave;
  if (row >= n) return;
  const float* p = in + (size_t)row * 128;
  float4 v = *(const float4*)(p + lane * 4);
  float s = v.x * v.x + v.y * v.y + v.z * v.z + v.w * v.w;
  for (int msk = 16; msk > 0; msk >>= 1) s += __shfl_xor(s, msk, 32);
  float d = 1.0f / fmaxf(sqrtf(s), 1e-12f);
  float4 o = {v.x * d, v.y * d, v.z * d, v.w * d};
  *(float4*)(out + (size_t)row * 128 + lane * 4) = o;
}

// ---------------------------------------------------------------------------
// Segment reductions over sorted segment ids (feature dim 128)
// ---------------------------------------------------------------------------
__device__ __forceinline__ int lower_bound_i(const int* a, int n, int key) {
  int lo = 0, hi = n;
  while (lo < hi) { int mid = (lo + hi) >> 1; if (a[mid] < key) lo = mid + 1; else hi = mid; }
  return lo;
}

__global__ void seg_max_k(const float* __restrict__ z, const float* __restrict__ noise,
                          const int* __restrict__ seg, int n,
                          float* __restrict__ zg, float* __restrict__ zpg)
{
  int g = blockIdx.x, c = threadIdx.x;  // 128 threads
  int s = lower_bound_i(seg, n, g);
  int e = lower_bound_i(seg, n, g + 1);
  float m1 = -INFINITY, m2 = -INFINITY;
  for (int r = s; r < e; r++) {
    float v = z[(size_t)r * 128 + c];
    m1 = fmaxf(m1, v);
    m2 = fmaxf(m2, v + noise[(size_t)r * 128 + c]);
  }
  zg[g * 128 + c] = m1;
  zpg[g * 128 + c] = m2;
}

__global__ void seg_mean_k(const float* __restrict__ z, const int* __restrict__ seg,
                           int n, float* __restrict__ out)
{
  int g = blockIdx.x, c = threadIdx.x;  // 128 threads
  int s = lower_bound_i(seg, n, g);
  int e = lower_bound_i(seg, n, g + 1);
  float acc = 0.0f;
  for (int r = s; r < e; r++) acc += z[(size_t)r * 128 + c];
  float cnt = (float)max(e - s, 1);
  out[g * 128 + c] = acc / cnt;
}

// ---------------------------------------------------------------------------
// Host-side orchestration
// ---------------------------------------------------------------------------
static inline int ceildiv_ll(long long a, int b) { return (int)((a + b - 1) / b); }

static void gemm(const float* A, const float* W, const float* bias, float* out,
                 int n, int K, int M, int act, hipStream_t st)
{
  dim3 grid(M >> 6, (unsigned)((n + 63) >> 6));
  gemm_wmma_k<<<grid, dim3(128), 0, st>>>(A, W, bias, out, n, K, M, act);
}

static void run_bn(const float* x, int n, int M, const float* g, const float* b,
                   float* bnsum /* 512 contiguous floats: sum|ssq */,
                   float* cA, float* cC, hipStream_t st)
{
  hipMemsetAsync(bnsum, 0, 512 * sizeof(float), st);
  int mshift = (M == 256) ? 8 : 7;
  const int rpb = 512;
  bn_stats_k<<<ceildiv_ll(n, rpb), 256, 0, st>>>(x, n, mshift, rpb, bnsum, bnsum + 256);
  bn_coef_k<<<1, 256, 0, st>>>(bnsum, bnsum + 256, g, b, M, 1.0f / (float)n, cA, cC);
}

struct BlkP {
  const float *Wc1, *bc1, *Wc2, *bc2, *Ws, *bs;
  const float *bn1g, *bn1b, *bn2g, *bn2b;
};

// One residual GCN block over the full N-node graph. OUT may alias B.
static void run_block(const float* X, int K, int M,
                      const int* src, const int* dst,
                      const float* dinv, const float* invd,
                      const BlkP& p,
                      float* A, float* B, float* R, float* OUT,
                      float* bnsum, float* cA, float* cC, hipStream_t st)
{
  const int mshift = (M == 256) ? 8 : 7;
  const int eshift = mshift - 2;
  const long long elems = (long long)NN << mshift;
  const int eb = ceildiv_ll(elems >> 2, 256);
  const int sb = ceildiv_ll(((long long)EE) << eshift, 256);

  gemm(X, p.Ws, p.bs, R, NN, K, M, 0, st);                       // residual path
  gemm(X, p.Wc1, nullptr, A, NN, K, M, 0, st);                   // h1 = X @ W1
  agg_init_k<<<eb, 256, 0, st>>>(A, invd, p.bc1, B, NN, mshift); // self loop + bias
  scatter_k<<<sb, 256, 0, st>>>(A, src, dst, dinv, B, EE, eshift, M);
  run_bn(B, NN, M, p.bn1g, p.bn1b, bnsum, cA, cC, st);
  bn_apply_k<<<eb, 256, 0, st>>>(B, cA, cC, nullptr, A, NN, mshift, 1); // relu(bn)
  gemm(A, p.Wc2, nullptr, B, NN, M, M, 0, st);                   // h2 = h @ W2
  agg_init_k<<<eb, 256, 0, st>>>(B, invd, p.bc2, A, NN, mshift);
  scatter_k<<<sb, 256, 0, st>>>(B, src, dst, dinv, A, EE, eshift, M);
  run_bn(A, NN, M, p.bn2g, p.bn2b, bnsum, cA, cC, st);
  bn_apply_k<<<eb, 256, 0, st>>>(A, cA, cC, R, OUT, NN, mshift, 1); // relu(bn + res)
}

extern "C" void kernel_launch(void* const* d_in, const int* in_sizes, int n_in,
                              void* d_out, int out_size, void* d_ws, size_t ws_size,
                              hipStream_t stream)
{
  (void)in_sizes; (void)n_in; (void)out_size; (void)ws_size;

  // ---- inputs (setup_inputs insertion order; params pytree-flattened) ----
  const float* x      = (const float*)d_in[0];
  const int*   ei     = (const int*)d_in[1];   // (2,E): [0:E)=src, [E:2E)=dst
  const int*   batch  = (const int*)d_in[2];
  const float* noise  = (const float*)d_in[3];
  const float* posx   = (const float*)d_in[4];
  const int*   pei    = (const int*)d_in[5];
  const int*   pbatch = (const int*)d_in[6];
  const float* tx     = (const float*)d_in[7];

  int idx = 8;
  // "dec"
  const float* dec0W = (const float*)d_in[idx++]; const float* dec0b = (const float*)d_in[idx++];
  const float* dec1W = (const float*)d_in[idx++]; const float* dec1b = (const float*)d_in[idx++];
  // block reader: keys sorted -> bn1(b,g), bn2(b,g), conv1(W,b), conv2(W,b), short(W,b)
  auto readBlk = [&](BlkP& p) {
    p.bn1b = (const float*)d_in[idx++]; p.bn1g = (const float*)d_in[idx++];
    p.bn2b = (const float*)d_in[idx++]; p.bn2g = (const float*)d_in[idx++];
    p.Wc1  = (const float*)d_in[idx++]; p.bc1  = (const float*)d_in[idx++];
    p.Wc2  = (const float*)d_in[idx++]; p.bc2  = (const float*)d_in[idx++];
    p.Ws   = (const float*)d_in[idx++]; p.bs   = (const float*)d_in[idx++];
  };
  // "enc"
  BlkP enc0, enc1, sub0, sub1;
  readBlk(enc0); readBlk(enc1);
  // "enc_node"
  const float* en0W = (const float*)d_in[idx++]; const float* en0b = (const float*)d_in[idx++];
  const float* en1W = (const float*)d_in[idx++]; const float* en1b = (const float*)d_in[idx++];
  // "enc_sub"
  readBlk(sub0); readBlk(sub1);
  // "proj"
  const float* pj0W = (const float*)d_in[idx++]; const float* pj0b = (const float*)d_in[idx++];
  const float* pj1W = (const float*)d_in[idx++]; const float* pj1b = (const float*)d_in[idx++];

  // ---- output layout (flat concat, reference return order) ----
  float* out_z   = (float*)d_out;                 // (N,128)
  float* out_rec = out_z   + (size_t)NN * F0C;    // (N,128)
  float* out_zg  = out_rec + (size_t)NN * F0C;    // (G,128)
  float* out_zpg = out_zg  + (size_t)GC * H1C;    // (G,128)
  float* out_psg = out_zpg + (size_t)GC * H1C;    // (GP,128)
  float* out_tz  = out_psg + (size_t)GPC * H1C;   // (G,128)

  // ---- workspace layout ----
  float* ws = (float*)d_ws;
  float* A = ws;                       // N*256
  float* B = A + (size_t)NN * 256;     // N*256
  float* R = B + (size_t)NN * 256;     // N*256
  float* sp = R + (size_t)NN * 256;
  float* dinvM = sp; sp += NN;         // main graph rsqrt(deg)
  float* invdM = sp; sp += NN;         // main graph 1/deg
  float* dinvP = sp; sp += NN;
  float* invdP = sp; sp += NN;
  float* bnsum = sp; sp += 512;        // sum(256) | ssq(256)
  float* cA    = sp; sp += 256;
  float* cC    = sp; sp += 256;
  float* zg    = sp; sp += (size_t)GC * H1C;
  float* zpg   = sp; sp += (size_t)GC * H1C;
  float* tH    = sp; sp += (size_t)GC * H0C;   // 500x256 scratch
  float* tH2   = sp; sp += (size_t)GC * H0C;

  const int nb  = ceildiv_ll(NN, 256);
  const int ebk = ceildiv_ll(EE, 256);

  // ---- degrees for both graphs ----
  fill_k<<<nb, 256, 0, stream>>>(dinvM, 1.0f, NN);
  deg_count_k<<<ebk, 256, 0, stream>>>(ei + EE, dinvM, EE);
  deg_fin_k<<<nb, 256, 0, stream>>>(dinvM, invdM, NN);
  fill_k<<<nb, 256, 0, stream>>>(dinvP, 1.0f, NN);
  deg_count_k<<<ebk, 256, 0, stream>>>(pei + EE, dinvP, EE);
  deg_fin_k<<<nb, 256, 0, stream>>>(dinvP, invdP, NN);

  // ---- main encoder: z = l2norm(block2(block1(x))) ----
  run_block(x, F0C, H0C, ei, ei + EE, dinvM, invdM, enc0, A, B, R, B, bnsum, cA, cC, stream);
  run_block(B, H0C, H1C, ei, ei + EE, dinvM, invdM, enc1, A, B, R, B, bnsum, cA, cC, stream);
  l2norm_k<<<ceildiv_ll(NN, 8), 256, 0, stream>>>(B, out_z, NN);

  // ---- decoder: x_recon = sigmoid(relu(z@W0+b0)@W1+b1) ----
  gemm(out_z, dec0W, dec0b, A, NN, H1C, H0C, 1, stream);
  gemm(A, dec1W, dec1b, out_rec, NN, H0C, F0C, 2, stream);

  // ---- graph readout + projection heads ----
  seg_max_k<<<GC, 128, 0, stream>>>(out_z, noise, batch, NN, zg, zpg);
  gemm(zg,  pj0W, pj0b, tH, GC, H1C, H1C, 1, stream);
  gemm(tH,  pj1W, pj1b, out_zg, GC, H1C, H1C, 0, stream);
  gemm(zpg, pj0W, pj0b, tH, GC, H1C, H1C, 1, stream);
  gemm(tH,  pj1W, pj1b, out_zpg, GC, H1C, H1C, 0, stream);

  // ---- subgraph encoder + segment mean ----
  run_block(posx, F0C, H0C, pei, pei + EE, dinvP, invdP, sub0, A, B, R, B, bnsum, cA, cC, stream);
  run_block(B,    H0C, H1C, pei, pei + EE, dinvP, invdP, sub1, A, B, R, B, bnsum, cA, cC, stream);
  l2norm_k<<<ceildiv_ll(NN, 8), 256, 0, stream>>>(B, B, NN);  // in place
  seg_mean_k<<<GPC, 128, 0, stream>>>(B, pbatch, NN, out_psg);

  // ---- node encoder: target_z ----
  gemm(tx, en0W, en0b, tH, GC, F0C, H0C, 1, stream);            // relu(lin0)
  run_bn(tH, GC, H0C, nullptr, nullptr, bnsum, cA, cC, stream); // bn with g=1,b=0
  bn_apply_k<<<ceildiv_ll(((long long)GC << 8) >> 2, 256), 256, 0, stream>>>(
      tH, cA, cC, nullptr, tH2, GC, 8, 0);                      // bn, no relu
  gemm(tH2, en1W, en1b, tH, GC, H0C, H1C, 0, stream);           // lin1
  l2norm_k<<<ceildiv_ll(GC, 8), 256, 0, stream>>>(tH, out_tz, GC);
}